// GCN1_13657996001612
// MI455X (gfx1250) — compile-verified
//
#include <hip/hip_runtime.h>

#define IN_DIM 128
#define OUT_DIM 64
#define LDS_PITCH 130  // 128 + 2 pad: spreads the 16 B-fragment columns across banks

typedef __attribute__((ext_vector_type(2))) float v2f;
typedef __attribute__((ext_vector_type(8))) float v8f;

// ---------------- utility: zero fill ----------------
__global__ __launch_bounds__(256) void zero_f32(float* __restrict__ p, long n) {
  long i = (long)blockIdx.x * blockDim.x + threadIdx.x;
  long stride = (long)gridDim.x * blockDim.x;
  for (; i < n; i += stride) p[i] = 0.0f;
}

// ---------------- degree at target nodes ----------------
__global__ __launch_bounds__(256) void degree_kernel(const long long* __restrict__ col,
                                                     float* __restrict__ deg, int n_edges) {
  int i = blockIdx.x * blockDim.x + threadIdx.x;
  if (i < n_edges) atomicAdd(&deg[col[i]], 1.0f);
}

// ---------------- deg^{-1/2} ----------------
__global__ __launch_bounds__(256) void dis_kernel(const float* __restrict__ deg,
                                                  float* __restrict__ dis, int n) {
  int i = blockIdx.x * blockDim.x + threadIdx.x;
  if (i < n) {
    float d = deg[i];
    dis[i] = (d > 0.0f) ? rsqrtf(fmaxf(d, 1e-12f)) : 0.0f;
  }
}

// ---------------- h = x @ W via V_WMMA_F32_16X16X4_F32 ----------------
// One wave -> 16 rows x 64 cols of h. W staged transposed in LDS so each
// B fragment (2 consecutive K values for one column) is one ds_load_b64.
__global__ __launch_bounds__(256) void gemm_wmma_f32(const float* __restrict__ x,
                                                     const float* __restrict__ W,
                                                     float* __restrict__ h, int n_nodes) {
  __shared__ float Wt[OUT_DIM * LDS_PITCH];  // ~33 KB of the 320 KB WGP LDS

  // cooperative transpose-stage: Wt[n][k] = W[k][n]
  for (int idx = threadIdx.x; idx < IN_DIM * OUT_DIM; idx += 256) {
    int k = idx >> 6;   // idx / OUT_DIM
    int n = idx & 63;   // idx % OUT_DIM
    Wt[n * LDS_PITCH + k] = W[idx];
  }
  __syncthreads();

  const int lane = threadIdx.x & 31;
  const int wave = threadIdx.x >> 5;
  const long row0 = ((long)blockIdx.x * 8 + wave) * 16;
  if (row0 >= (long)n_nodes) return;  // wave-uniform exit; EXEC stays all-ones for WMMA

  const int m     = lane & 15;   // A-matrix row / B,C,D column within tile
  const int half  = lane >> 4;   // 0: K pair {0,1}, 1: K pair {2,3}
  const int khalf = half * 2;

  // clamp A row for (hypothetical) tail tiles; reads duplicate data, store is masked
  long arow = row0 + m;
  if (arow >= n_nodes) arow = n_nodes - 1;
  const float* xrow = x + arow * IN_DIM;

  v8f acc0 = {}, acc1 = {}, acc2 = {}, acc3 = {};

  for (int k = 0; k < IN_DIM; k += 4) {
    v2f a  = *(const v2f*)(xrow + k + khalf);                              // A[m][k+khalf .. +1]
    v2f b0 = *(const v2f*)&Wt[( 0 + m) * LDS_PITCH + k + khalf];           // B[k..][n= 0+m]
    v2f b1 = *(const v2f*)&Wt[(16 + m) * LDS_PITCH + k + khalf];
    v2f b2 = *(const v2f*)&Wt[(32 + m) * LDS_PITCH + k + khalf];
    v2f b3 = *(const v2f*)&Wt[(48 + m) * LDS_PITCH + k + khalf];
    acc0 = __builtin_amdgcn_wmma_f32_16x16x4_f32(false, a, false, b0, (short)0, acc0, false, false);
    acc1 = __builtin_amdgcn_wmma_f32_16x16x4_f32(false, a, false, b1, (short)0, acc1, false, false);
    acc2 = __builtin_amdgcn_wmma_f32_16x16x4_f32(false, a, false, b2, (short)0, acc2, false, false);
    acc3 = __builtin_amdgcn_wmma_f32_16x16x4_f32(false, a, false, b3, (short)0, acc3, false, false);
  }

  // C/D layout: lane holds rows M = v + 8*half (v=0..7), column N = m (+16 per tile)
  const int drow = half * 8;
#pragma unroll
  for (int v = 0; v < 8; ++v) {
    long grow = row0 + drow + v;
    if (grow < n_nodes) {
      float* hb = h + grow * OUT_DIM + m;
      hb[0]  = acc0[v];
      hb[16] = acc1[v];
      hb[32] = acc2[v];
      hb[48] = acc3[v];
    }
  }
}

// ---------------- edge gather/scale/scatter (L2-resident) ----------------
// 16 threads per edge, each moving a float4 (64 dims total).
__global__ __launch_bounds__(256) void scatter_kernel(const float* __restrict__ h,
                                                      const long long* __restrict__ row,
                                                      const long long* __restrict__ col,
                                                      const float* __restrict__ dis,
                                                      float* __restrict__ out, int n_edges) {
  long gid = (long)blockIdx.x * blockDim.x + threadIdx.x;
  long e = gid >> 4;
  if (e >= n_edges) return;
  int sub = (int)(gid & 15) * 4;
  long r = row[e], c = col[e];
  float nrm = dis[r] * dis[c];
  float4 v = *(const float4*)(h + r * OUT_DIM + sub);
  float* dst = out + c * OUT_DIM + sub;
  atomicAdd(dst + 0, v.x * nrm);
  atomicAdd(dst + 1, v.y * nrm);
  atomicAdd(dst + 2, v.z * nrm);
  atomicAdd(dst + 3, v.w * nrm);
}

// ---------------- bias + ReLU, in place on d_out ----------------
__global__ __launch_bounds__(256) void bias_relu(float* __restrict__ out,
                                                 const float* __restrict__ b, long n_total) {
  long i = (long)blockIdx.x * blockDim.x + threadIdx.x;
  if (i < n_total) {
    float v = out[i] + b[i & 63];
    out[i] = v > 0.0f ? v : 0.0f;
  }
}

extern "C" void kernel_launch(void* const* d_in, const int* in_sizes, int n_in,
                              void* d_out, int out_size, void* d_ws, size_t ws_size,
                              hipStream_t stream) {
  const float* x      = (const float*)d_in[0];
  const long long* ei = (const long long*)d_in[1];  // int64 edge_index [2, E]
  const float* W      = (const float*)d_in[2];
  const float* b      = (const float*)d_in[3];
  float* out          = (float*)d_out;

  const int n_nodes = in_sizes[0] / IN_DIM;
  const int n_edges = in_sizes[1] / 2;
  const long long* row = ei;            // edge_index[0]
  const long long* col = ei + n_edges;  // edge_index[1]

  // workspace: deg[N] | dis[N] | h[N*64]
  float* deg = (float*)d_ws;
  float* dis = deg + n_nodes;
  float* h   = dis + n_nodes;

  zero_f32<<<512, 256, 0, stream>>>(deg, (long)n_nodes);
  zero_f32<<<2048, 256, 0, stream>>>(out, (long)n_nodes * OUT_DIM);

  degree_kernel<<<(n_edges + 255) / 256, 256, 0, stream>>>(col, deg, n_edges);
  dis_kernel<<<(n_nodes + 255) / 256, 256, 0, stream>>>(deg, dis, n_nodes);

  int row_tiles = (n_nodes + 15) / 16;
  int gemm_blocks = (row_tiles + 7) / 8;  // 8 waves (tiles) per 256-thread block
  gemm_wmma_f32<<<gemm_blocks, 256, 0, stream>>>(x, W, h, n_nodes);

  long sc_threads = (long)n_edges * 16;
  scatter_kernel<<<(int)((sc_threads + 255) / 256), 256, 0, stream>>>(h, row, col, dis, out, n_edges);

  long n_total = (long)n_nodes * OUT_DIM;
  bias_relu<<<(int)((n_total + 255) / 256), 256, 0, stream>>>(out, b, n_total);
}